// GAT_40621800685951
// MI455X (gfx1250) — compile-verified
//
#include <hip/hip_runtime.h>
#include <math.h>

typedef __attribute__((ext_vector_type(2))) float v2f;
typedef __attribute__((ext_vector_type(8))) float v8f;

#define IN_DIM 256
#define H_DIM  64

// ---------------------------------------------------------------- utilities
__global__ void k_zero(float* __restrict__ p, long long n) {
    long long i = (long long)blockIdx.x * blockDim.x + threadIdx.x;
    long long stride = (long long)gridDim.x * blockDim.x;
    for (; i < n; i += stride) p[i] = 0.0f;
}

__global__ void k_elu_inplace(float* __restrict__ p, long long n) {
    long long i = (long long)blockIdx.x * blockDim.x + threadIdx.x;
    long long stride = (long long)gridDim.x * blockDim.x;
    for (; i < n; i += stride) {
        float v = p[i];
        p[i] = v > 0.0f ? v : (expf(v) - 1.0f);
    }
}

// ---------------------------------------------------------------- GEMM (WMMA f32)
// out[m, n] = sum_k X[m,k] * W[k,n] + bias[n]  (+ addin[m,n])  (optional ELU)
// X: [nrows, 256], W: [256, 64] row-major, out: [nrows, ldo].
// One wave computes a 16-row x 64-col strip using V_WMMA_F32_16X16X4_F32.
// W is staged into LDS K-pair-interleaved so each lane's B fragment
// {W[k][n], W[k+1][n]} is one aligned ds_load_b64 into an even VGPR pair.
__global__ __launch_bounds__(256)
void k_gemm(const float* __restrict__ X, int ldx,
            const float* __restrict__ W,
            const float* __restrict__ bias,
            const float* __restrict__ addin, int lda,
            float* __restrict__ out, int ldo,
            int nrows, int do_elu)
{
    // Ws2[p*64 + n] = { W[2p][n], W[2p+1][n] }   (128 pairs x 64 cols = 64 KB)
    __shared__ v2f Ws2[(IN_DIM / 2) * H_DIM];
    for (int i = threadIdx.x; i < (IN_DIM / 2) * H_DIM; i += blockDim.x) {
        const int p = i >> 6;        // K-pair index
        const int n = i & 63;        // column
        v2f t;
        t.x = W[(2 * p)     * H_DIM + n];
        t.y = W[(2 * p + 1) * H_DIM + n];
        Ws2[i] = t;
    }
    __syncthreads();

    const int lane  = threadIdx.x & 31;
    const int wave  = threadIdx.x >> 5;
    const int strip = blockIdx.x * (blockDim.x >> 5) + wave;
    const int row0  = strip * 16;
    if (row0 >= nrows) return;

    const int half = lane >> 4;   // 0: lanes 0-15, 1: lanes 16-31
    const int l15  = lane & 15;

    v8f zero8 = {};
    v8f acc[4] = { zero8, zero8, zero8, zero8 };

    // A fragment 16x4 f32 layout: lane<16 -> M=lane, {v0,v1}={K0,K1};
    //                             lane>=16 -> M=lane-16, {v0,v1}={K2,K3}
    const float* xrow = X + (size_t)(row0 + l15) * ldx + 2 * half;

    for (int k = 0; k < IN_DIM; k += 4) {
        v2f a = *(const v2f*)(xrow + k);               // contiguous {K, K+1}
        // B fragment 4x16 f32: lane<16 -> N=lane, {v0,v1}={K0,K1};
        //                      lane>=16 -> N=lane-16, {v0,v1}={K2,K3}
        const v2f* wp = Ws2 + ((k >> 1) + half) * H_DIM + l15;
        #pragma unroll
        for (int nt = 0; nt < 4; ++nt) {
            v2f b = wp[nt * 16];                       // one ds_load_b64
            acc[nt] = __builtin_amdgcn_wmma_f32_16x16x4_f32(
                false, a, false, b, (short)0, acc[nt], false, false);
        }
    }

    // C/D layout: lane<16 holds M=r (vgpr r), lane>=16 holds M=8+r; N=lane&15
    #pragma unroll
    for (int r = 0; r < 8; ++r) {
        const int m = row0 + r + 8 * half;
        #pragma unroll
        for (int nt = 0; nt < 4; ++nt) {
            const int n = l15 + nt * 16;
            float v = acc[nt][r] + bias[n];
            if (addin) v += addin[(size_t)m * lda + n];
            if (do_elu) v = v > 0.0f ? v : (expf(v) - 1.0f);
            out[(size_t)m * ldo + n] = v;
        }
    }
}

// ---------------------------------------------------------------- attention coefs
// a1[i] = ft[i,:] . wl + bl ; a2[i] = ft[i,:] . wr + br  (one wave32 per node)
__global__ __launch_bounds__(256)
void k_attn(const float* __restrict__ ft,
            const float* __restrict__ wl, const float* __restrict__ wr,
            const float* __restrict__ bl, const float* __restrict__ br,
            float* __restrict__ a1, float* __restrict__ a2, int nnodes)
{
    const int lane = threadIdx.x & 31;
    const int wave = threadIdx.x >> 5;
    const int i = blockIdx.x * (blockDim.x >> 5) + wave;
    if (i >= nnodes) return;

    const float* f = ft + (size_t)i * H_DIM;
    float f0 = f[lane], f1 = f[lane + 32];
    float s1 = f0 * wl[lane] + f1 * wl[lane + 32];
    float s2 = f0 * wr[lane] + f1 * wr[lane + 32];
    #pragma unroll
    for (int off = 16; off > 0; off >>= 1) {
        s1 += __shfl_xor(s1, off, 32);
        s2 += __shfl_xor(s2, off, 32);
    }
    if (lane == 0) {
        a1[i] = s1 + bl[0];
        a2[i] = s2 + br[0];
    }
}

// ---------------------------------------------------------------- edge passes
// Pass 1: score = leaky_relu(a1[dst] + a2[src], 0.01); per-dst max (clamped at 0).
// amax pre-zeroed; positive floats order-isomorphic to their int bits.
__global__ void k_edge_max(const int* __restrict__ src, const int* __restrict__ dst,
                           const float* __restrict__ a1, const float* __restrict__ a2,
                           float* __restrict__ sc, float* __restrict__ amax, int E)
{
    int e = blockIdx.x * blockDim.x + threadIdx.x;
    if (e >= E) return;
    int d = dst[e];
    float x = a1[d] + a2[src[e]];
    float s = x > 0.0f ? x : 0.01f * x;
    sc[e] = s;
    if (s > 0.0f)
        atomicMax((int*)amax + d, __float_as_int(s));
}

// Pass 2: ea = exp(s - amax[dst]); asum[src] += ea (in-place overwrite of sc).
__global__ void k_edge_exp(const int* __restrict__ src, const int* __restrict__ dst,
                           float* __restrict__ sc, const float* __restrict__ amax,
                           float* __restrict__ asum, int E)
{
    int e = blockIdx.x * blockDim.x + threadIdx.x;
    if (e >= E) return;
    float ea = expf(sc[e] - amax[dst[e]]);
    sc[e] = ea;
    atomicAdd(asum + src[e], ea);
}

// Pass 3: out[src, col0:col0+64] += (ea/asum[dst]) * ft[dst, :]  (wave32 per edge)
__global__ __launch_bounds__(256)
void k_spmm(const int* __restrict__ src, const int* __restrict__ dst,
            const float* __restrict__ ea, const float* __restrict__ asum,
            const float* __restrict__ ft,
            float* __restrict__ out, int ldo, int col0, int E)
{
    const int lane = threadIdx.x & 31;
    const int wave = threadIdx.x >> 5;
    const int e = blockIdx.x * (blockDim.x >> 5) + wave;
    if (e >= E) return;

    const int s = src[e], d = dst[e];
    const float attn = ea[e] / asum[d];
    const float* f = ft + (size_t)d * H_DIM;
    float* o = out + (size_t)s * ldo + col0;
    atomicAdd(&o[lane],      attn * f[lane]);
    atomicAdd(&o[lane + 32], attn * f[lane + 32]);
}

// ---------------------------------------------------------------- driver
extern "C" void kernel_launch(void* const* d_in, const int* in_sizes, int n_in,
                              void* d_out, int out_size, void* d_ws, size_t ws_size,
                              hipStream_t stream)
{
    const float* features = (const float*)d_in[0];
    const int*   src      = (const int*)d_in[1];
    const int*   dst      = (const int*)d_in[2];
    const float* W_fc     = (const float*)d_in[3];
    const float* b_fc     = (const float*)d_in[4];
    const float* w_al     = (const float*)d_in[5];
    const float* b_al     = (const float*)d_in[6];
    const float* w_ar     = (const float*)d_in[7];
    const float* b_ar     = (const float*)d_in[8];
    const float* W_res    = (const float*)d_in[9];
    const float* b_res    = (const float*)d_in[10];

    const int N = in_sizes[0] / IN_DIM;
    const int E = in_sizes[1];

    float* ws     = (float*)d_ws;
    float* last   = ws;  ws += (size_t)N * IN_DIM;   // [N,256] concat of 4 heads
    float* ft     = ws;  ws += (size_t)N * H_DIM;    // [N,64] current head transform
    float* out5   = ws;  ws += (size_t)N * H_DIM;    // [N,64] final head output
    float* a1     = ws;  ws += N;
    float* a2     = ws;  ws += N;
    float* amax   = ws;  ws += N;                    // amax & asum contiguous
    float* asum   = ws;  ws += N;
    float* escore = ws;  ws += E;                    // per-edge score / ea

    const int zgrid      = 2048;
    const int gemm_grid  = ((N + 15) / 16 + 7) / 8;  // 8 waves/block, 16 rows/wave
    const int node_wgrid = (N + 7) / 8;              // wave per node
    const int edge_tgrid = (E + 255) / 256;          // thread per edge
    const int edge_wgrid = (E + 7) / 8;              // wave per edge

    k_zero<<<zgrid, 256, 0, stream>>>(last, (long long)N * IN_DIM);

    for (int h = 0; h < 5; ++h) {
        const float* X  = (h < 4) ? features : last;
        const float* W  = W_fc + (size_t)h * IN_DIM * H_DIM;
        const float* bf = b_fc + (size_t)h * H_DIM;

        k_gemm<<<gemm_grid, 256, 0, stream>>>(X, IN_DIM, W, bf,
                                              (const float*)nullptr, 0,
                                              ft, H_DIM, N, 0);
        k_attn<<<node_wgrid, 256, 0, stream>>>(ft,
                                               w_al + (size_t)h * H_DIM,
                                               w_ar + (size_t)h * H_DIM,
                                               b_al + h, b_ar + h, a1, a2, N);
        k_zero<<<zgrid, 256, 0, stream>>>(amax, (long long)2 * N);  // amax + asum
        k_edge_max<<<edge_tgrid, 256, 0, stream>>>(src, dst, a1, a2, escore, amax, E);
        k_edge_exp<<<edge_tgrid, 256, 0, stream>>>(src, dst, escore, amax, asum, E);

        if (h < 4) {
            k_spmm<<<edge_wgrid, 256, 0, stream>>>(src, dst, escore, asum, ft,
                                                   last, IN_DIM, h * H_DIM, E);
            if (h == 3)  // all 4 heads accumulated -> elementwise ELU on concat
                k_elu_inplace<<<zgrid, 256, 0, stream>>>(last, (long long)N * IN_DIM);
        } else {
            k_zero<<<zgrid, 256, 0, stream>>>(out5, (long long)N * H_DIM);
            k_spmm<<<edge_wgrid, 256, 0, stream>>>(src, dst, escore, asum, ft,
                                                   out5, H_DIM, 0, E);
        }
    }

    // d_out = elu(last @ W_res + b_res + out5)
    k_gemm<<<gemm_grid, 256, 0, stream>>>(last, IN_DIM, W_res, b_res,
                                          out5, H_DIM,
                                          (float*)d_out, H_DIM, N, 1);
    (void)n_in; (void)out_size; (void)ws_size;
}